// GAT_70841190580931
// MI455X (gfx1250) — compile-verified
//
#include <hip/hip_runtime.h>
#include <hip/hip_bf16.h>

// Viterbi CRF decode, B=1024, T=1024, K=48, START=46, STOP=47, NEG=-10000
// Forward: 1 wave32 per sequence; transitions in VGPRs; fv broadcast via v_readlane;
//          feats streamed to LDS by the Tensor Data Mover (double buffered, ds_load reads).
// Backtrack: 1 thread per sequence; full-row loads + branchless byte select.

#define KTAGS 48
#define TLEN  1024
#define BATCH 1024
#define START_TAG 46
#define STOP_TAG  47
#define NEGV  (-10000.0f)
#define TILE  16
#define NTILES (TLEN / TILE)
#define WPB   8   // waves per block (256 threads)

typedef unsigned int  v4u __attribute__((ext_vector_type(4)));
typedef int           v8i __attribute__((ext_vector_type(8)));
typedef int           v4i __attribute__((ext_vector_type(4)));

__device__ __forceinline__ float rdlane(float v, int lane) {
    return __uint_as_float((unsigned)__builtin_amdgcn_readlane((int)__float_as_uint(v), lane));
}

// Issue a TDM load of a TILE x KTAGS f32 tile (row-major, stride KTAGS) into LDS.
__device__ __forceinline__ void tdm_load_tile(const float* gsrc, unsigned lds_addr) {
    unsigned long long ga = (unsigned long long)(const void*)gsrc;
    // D# group0: count=1 | lds_addr | global_addr[31:0] | global_addr[56:32] + type=2
    v4u g0 = { 1u,
               lds_addr,
               (unsigned)ga,
               ((unsigned)(ga >> 32) & 0x1FFFFFFu) | 0x80000000u };
    // D# group1:
    //  d0: workgroup_mask=0, data_size=2 (4B)            -> 0x20000
    //  d1: atomic_barrier_addr=0, tensor_dim0[15:0]=48   -> 48<<16
    //  d2: tensor_dim0[31:16]=0, tensor_dim1[15:0]=0     -> 0       (tensor_dim1 = 1<<20 rows)
    //  d3: tensor_dim1[31:16]=0x10, tile_dim0=48         -> (48<<16)|0x10
    //  d4: tile_dim1=TILE, tile_dim2=0                   -> TILE
    //  d5: tensor_dim0_stride[31:0]=48
    //  d6: stride0[47:32]=0, tensor_dim1_stride[15:0]=0  -> 0
    //  d7: 0
    v8i g1 = { 0x20000, (KTAGS << 16), 0, (KTAGS << 16) | 0x10, TILE, KTAGS, 0, 0 };
    v4i z4 = { 0, 0, 0, 0 };
    v8i z8 = { 0, 0, 0, 0, 0, 0, 0, 0 };
    __builtin_amdgcn_tensor_load_to_lds(g0, g1, z4, z4, z8, 0);
}

__shared__ float featBuf[WPB][2][TILE * KTAGS];   // 48 KB

// One 16-step tile of the Viterbi recurrence, reading feats from LDS (ds_load path).
__device__ __forceinline__ void run_tile(int w, int buf, int lane, int tile,
                                         const float tA[KTAGS], const float tB[KTAGS],
                                         float& fvA, float& fvB,
                                         unsigned char* __restrict__ bp, size_t bpBase) {
    asm volatile("" ::: "memory");   // order LDS reads after s_wait_tensorcnt
#pragma unroll 1
    for (int r = 0; r < TILE; ++r) {
        const int t = tile * TILE + r;
        // direct __shared__ access -> addrspace(3) -> ds_load_b32
        const float featA = featBuf[w][buf][r * KTAGS + lane];
        const float featB = featBuf[w][buf][r * KTAGS + (lane & 15) + 32];

        float bestA, bestB; int idxA, idxB;
#pragma unroll
        for (int p = 0; p < KTAGS; ++p) {
            const float sfv = (p < 32) ? rdlane(fvA, p) : rdlane(fvB, p - 32);
            const float sA = sfv + tA[p];
            const float sB = sfv + tB[p];
            if (p == 0) {
                bestA = sA; idxA = 0; bestB = sB; idxB = 0;
            } else {
                const bool gA = sA > bestA;          // strict > => first-occurrence argmax
                bestA = gA ? sA : bestA; idxA = gA ? p : idxA;
                const bool gB = sB > bestB;
                bestB = gB ? sB : bestB; idxB = gB ? p : idxB;
            }
        }
        // store backpointers (bytes)
        unsigned char* row = bp + bpBase + (size_t)t * KTAGS;
        row[lane] = (unsigned char)idxA;
        if (lane < 16) row[lane + 32] = (unsigned char)idxB;

        fvA = bestA + featA;
        fvB = bestB + featB;
    }
}

__global__ __launch_bounds__(256)
void viterbi_forward(const float* __restrict__ feats, const float* __restrict__ trans,
                     float* __restrict__ out_scores, unsigned char* __restrict__ bp,
                     int* __restrict__ startTag) {
    const int lane = threadIdx.x & 31;
    const int w    = threadIdx.x >> 5;
    const int b    = blockIdx.x * WPB + w;

    // ---- preload transition rows into registers: lane owns rows `lane` and `lane+32` ----
    float tA[KTAGS], tB[KTAGS];
    const int rowA = lane;
    const int rowB = (lane < 16) ? (lane + 32) : lane;   // clamp: lanes>=16 B-half unused
#pragma unroll
    for (int q = 0; q < KTAGS / 4; ++q) {
        float4 va = *(const float4*)(trans + (size_t)rowA * KTAGS + q * 4);
        float4 vb = *(const float4*)(trans + (size_t)rowB * KTAGS + q * 4);
        tA[q*4+0] = va.x; tA[q*4+1] = va.y; tA[q*4+2] = va.z; tA[q*4+3] = va.w;
        tB[q*4+0] = vb.x; tB[q*4+1] = vb.y; tB[q*4+2] = vb.z; tB[q*4+3] = vb.w;
    }
    const int c47 = (lane < 16) ? (32 + lane) : lane;
    const float t47A = trans[STOP_TAG * KTAGS + lane];
    const float t47B = trans[STOP_TAG * KTAGS + c47];

    // ---- init forward variable (distributed: lane l holds tags l and l+32) ----
    float fvA = NEGV;
    float fvB = (lane == (START_TAG - 32)) ? 0.0f : NEGV;

    // ---- prime TDM double buffer ----
    const float* fbase = feats + (size_t)b * TLEN * KTAGS;
    unsigned lds0 = (unsigned)(unsigned long long)(const void*)&featBuf[w][0][0];
    unsigned lds1 = (unsigned)(unsigned long long)(const void*)&featBuf[w][1][0];
    tdm_load_tile(fbase + 0 * TILE * KTAGS, lds0);
    tdm_load_tile(fbase + 1 * TILE * KTAGS, lds1);

    const size_t bpBase = (size_t)b * TLEN * KTAGS;

    // steady state: one outstanding prefetch while processing current tile
#pragma unroll 1
    for (int tile = 0; tile < NTILES - 1; ++tile) {
        const int buf = tile & 1;
        __builtin_amdgcn_s_wait_tensorcnt(1);   // current tile's transfer complete
        run_tile(w, buf, lane, tile, tA, tB, fvA, fvB, bp, bpBase);
        if (tile + 2 < NTILES)
            tdm_load_tile(fbase + (size_t)(tile + 2) * TILE * KTAGS, buf ? lds1 : lds0);
    }
    // last tile
    __builtin_amdgcn_s_wait_tensorcnt(0);
    run_tile(w, (NTILES - 1) & 1, lane, NTILES - 1, tA, tB, fvA, fvB, bp, bpBase);

    // ---- terminal: fv + trans[STOP], argmax over 48 tags ----
    const float termA = fvA + t47A;
    const float termB = fvB + t47B;
    float best; int bi;
#pragma unroll
    for (int p = 0; p < KTAGS; ++p) {
        const float v = (p < 32) ? rdlane(termA, p) : rdlane(termB, p - 32);
        if (p == 0) { best = v; bi = 0; }
        else { const bool g = v > best; best = g ? v : best; bi = g ? p : bi; }
    }
    if (lane == 0) {
        out_scores[b] = best;
        startTag[b]   = bi;
    }
}

__global__ __launch_bounds__(256)
void viterbi_backtrack(const unsigned char* __restrict__ bp, const int* __restrict__ startTag,
                       float* __restrict__ pathOut) {
    const int b = blockIdx.x * blockDim.x + threadIdx.x;
    if (b >= BATCH) return;

    int tag = startTag[b];
    const unsigned char* base = bp + (size_t)b * TLEN * KTAGS;
    float* po = pathOut + (size_t)b * TLEN;

#pragma unroll 4
    for (int t = TLEN - 1; t >= 0; --t) {
        po[t] = (float)tag;
        // load full 48-byte row (address independent of tag -> pipelineable)
        const uint4* rp = (const uint4*)(base + (size_t)t * KTAGS);
        const uint4 r0 = rp[0], r1 = rp[1], r2 = rp[2];
        const int wi = tag >> 2;
        const uint4 rs = (wi < 4) ? r0 : ((wi < 8) ? r1 : r2);
        const int wj = wi & 3;
        const unsigned wsel = (wj & 2) ? ((wj & 1) ? rs.w : rs.z)
                                       : ((wj & 1) ? rs.y : rs.x);
        tag = (int)((wsel >> ((tag & 3) * 8)) & 0xFFu);
    }
}

extern "C" void kernel_launch(void* const* d_in, const int* in_sizes, int n_in,
                              void* d_out, int out_size, void* d_ws, size_t ws_size,
                              hipStream_t stream) {
    const float* feats = (const float*)d_in[0];   // [B, T, K] f32
    const float* trans = (const float*)d_in[1];   // [K, K]    f32
    float* out = (float*)d_out;                   // [B] scores ++ [B*T] path (as float)

    unsigned char* bp = (unsigned char*)d_ws;                             // B*T*K bytes
    int* startTag = (int*)((char*)d_ws + (size_t)BATCH * TLEN * KTAGS);   // B ints

    viterbi_forward<<<BATCH / WPB, 256, 0, stream>>>(feats, trans, out, bp, startTag);
    viterbi_backtrack<<<(BATCH + 255) / 256, 256, 0, stream>>>(bp, startTag, out + BATCH);
}